// BiMambav3_31971736552179
// MI455X (gfx1250) — compile-verified
//
#include <hip/hip_runtime.h>
#include <hip/hip_bf16.h>

// BiMamba forward for MI455X (gfx1250, wave32).
// Dense projections: V_WMMA_F32_16X16X4_F32 (full f32 precision) with
// block-shared activation tiles staged into LDS via GLOBAL_LOAD_ASYNC_TO_LDS
// (tracked by ASYNCcnt), weights streamed with global_prefetch_b8.
// Selective scan is register-resident: one SSM state per lane, 16 lanes per
// channel, dt-projection fused in.

#define BATCH   2
#define LSEQ    1024
#define DMODEL  768
#define DSTATE  16
#define DINNER  1536
#define DTRANK  48
#define XDBLR   160   // rows of x_dbl = 2*(DTRANK + 2*DSTATE)

typedef __attribute__((ext_vector_type(2))) float v2f;
typedef __attribute__((ext_vector_type(8))) float v8f;

#if __has_builtin(__builtin_amdgcn_global_load_async_to_lds_b128) && \
    __has_builtin(__builtin_amdgcn_global_load_async_to_lds_b32)  && \
    __has_builtin(__builtin_amdgcn_s_wait_asynccnt)
#define HAVE_ASYNC_LDS 1
#endif

#ifdef HAVE_ASYNC_LDS
typedef __attribute__((ext_vector_type(4))) int i32x4v;
// Builtin signatures (from compiler diagnostics):
//   b128: (int4 AS1*, int4 AS3*, imm offset, imm cpol)
//   b32 : (int  AS1*, int  AS3*, imm offset, imm cpol)
__device__ __forceinline__ void async_cp16(const float* g, float* l) {
  __builtin_amdgcn_global_load_async_to_lds_b128(
      (__attribute__((address_space(1))) i32x4v*)(void*)g,
      (__attribute__((address_space(3))) i32x4v*)(void*)l, 0, 0);
}
__device__ __forceinline__ void async_cp4(const float* g, float* l) {
  __builtin_amdgcn_global_load_async_to_lds_b32(
      (__attribute__((address_space(1))) int*)(void*)g,
      (__attribute__((address_space(3))) int*)(void*)l, 0, 0);
}
__device__ __forceinline__ void async_wait0() {
  __builtin_amdgcn_s_wait_asynccnt(0);
}
#endif

__device__ __forceinline__ v8f wmma4(v2f a, v2f b, v8f c) {
  // D = A(16x4,f32) * B(4x16,f32) + C(16x16,f32)
  return __builtin_amdgcn_wmma_f32_16x16x4_f32(
      /*neg_a=*/false, a, /*neg_b=*/false, b,
      /*c_mod=*/(short)0, c, /*reuse_a=*/false, /*reuse_b=*/false);
}

__device__ __forceinline__ float silu_f(float v) {
  return v / (1.0f + __expf(-v));
}

// ---------------------------------------------------------------------------
// K1: xz[b,e,l] = sum_d x[b,l,d] * W[e,d]
// Block = 8 waves sharing one 64-column activation tile staged in LDS
// ([n][k], k-stride padded to 68 dwords -> conflict-free ds_load_b64).
// grid (LSEQ/64, 3072/128, B), block 256.
// ---------------------------------------------------------------------------
__global__ __launch_bounds__(256)
void k_in_proj(const float* __restrict__ x, const float* __restrict__ w,
               float* __restrict__ xz) {
  __shared__ float bs[64 * 68];
  const int tid  = threadIdx.x;
  const int lane = tid & 31;
  const int wave = tid >> 5;
  const int lm   = lane & 15;
  const int half = lane >> 4;
  const int kh   = half * 2;
  const int b  = blockIdx.z;
  const int n0 = blockIdx.x * 64;
  const int m0 = blockIdx.y * 128 + wave * 16;

  const float* xb   = x + (size_t)b * LSEQ * DMODEL;
  const float* arow = w + (size_t)(m0 + lm) * DMODEL + kh;
  const int snl = tid >> 2;         // staged row n (0..63)
  const int skq = (tid & 3) * 16;   // 16-float k chunk

  v8f acc[4] = {};
  for (int ks = 0; ks < DMODEL; ks += 64) {
    const float* gsrc = xb + (size_t)(n0 + snl) * DMODEL + ks + skq;
    float* ldst = &bs[snl * 68 + skq];
#ifdef HAVE_ASYNC_LDS
#pragma unroll
    for (int i = 0; i < 4; ++i) async_cp16(gsrc + i * 4, ldst + i * 4);
    async_wait0();
#else
#pragma unroll
    for (int i = 0; i < 4; ++i)
      *(float4*)(ldst + i * 4) = *(const float4*)(gsrc + i * 4);
#endif
    __syncthreads();
#pragma unroll
    for (int kk = 0; kk < 64; kk += 4) {
      if ((kk & 31) == 0) __builtin_prefetch(arow + 64, 0, 3);
      v2f a; a.x = arow[0]; a.y = arow[1]; arow += 4;
      const float* bb = &bs[lm * 68 + kk + kh];
#pragma unroll
      for (int j = 0; j < 4; ++j) {
        v2f bf; bf.x = bb[j * 16 * 68]; bf.y = bb[j * 16 * 68 + 1];
        acc[j] = wmma4(a, bf, acc[j]);
      }
    }
    __syncthreads();
  }
  float* outb = xz + (size_t)b * (2 * DINNER) * LSEQ;
#pragma unroll
  for (int j = 0; j < 4; ++j)
#pragma unroll
    for (int r = 0; r < 8; ++r)
      outb[(size_t)(m0 + r + half * 8) * LSEQ + n0 + j * 16 + lm] = acc[j][r];
}

// ---------------------------------------------------------------------------
// K2: dense conv1d (k=3, same-pad) as 3 accumulated GEMMs + bias + SiLU.
// The staged tile carries a 2-column halo so all 3 taps read the same LDS
// slab (3x global traffic saved). The stage is a transpose (global rows are
// k-major, LDS is [n][k]) -> per-lane-addressed async b32 copies. Boundary
// column blocks use a sync zero-filling stage.
// grid (LSEQ/64, DINNER/128, B), block 256.
// ---------------------------------------------------------------------------
__global__ __launch_bounds__(256)
void k_conv(const float* __restrict__ xz, const float* __restrict__ wc,
            const float* __restrict__ bc, float* __restrict__ xi) {
  __shared__ float bs[68 * 68];     // [nl][kl], nl = n - (n0-2), stride 68
  const int tid  = threadIdx.x;
  const int lane = tid & 31;
  const int wave = tid >> 5;
  const int lm   = lane & 15;
  const int half = lane >> 4;
  const int kh   = half * 2;
  const int b  = blockIdx.z;
  const int n0 = blockIdx.x * 64;
  const int m0 = blockIdx.y * 128 + wave * 16;

  const float* src = xz + (size_t)b * (2 * DINNER) * LSEQ;  // rows 0..1535 = pre-conv xi
  const float* ap  = wc + (size_t)(m0 + lm) * (DINNER * 3); // stride-3 along K per tap
  const int skl = tid >> 2;          // staged k row (0..63)
  const int snb = (tid & 3) * 17;    // 17-column n chunk
  const bool interior = (n0 >= 2) && (n0 + 66 <= LSEQ);

  v8f acc[4] = {};
  for (int ks = 0; ks < DINNER; ks += 64) {
    const float* grow = src + (size_t)(ks + skl) * LSEQ + (n0 - 2);
    if (interior) {
#ifdef HAVE_ASYNC_LDS
#pragma unroll
      for (int i = 0; i < 17; ++i) {
        const int nl = snb + i;
        async_cp4(grow + nl, &bs[nl * 68 + skl]);   // transposing stage
      }
      async_wait0();
#else
#pragma unroll
      for (int i = 0; i < 17; ++i) {
        const int nl = snb + i;
        bs[nl * 68 + skl] = grow[nl];
      }
#endif
    } else {
#pragma unroll
      for (int i = 0; i < 17; ++i) {
        const int nl = snb + i;
        const int n  = n0 - 2 + nl;
        bs[nl * 68 + skl] =
            ((unsigned)n < (unsigned)LSEQ) ? src[(size_t)(ks + skl) * LSEQ + n] : 0.0f;
      }
    }
    __syncthreads();
#pragma unroll
    for (int kk = 0; kk < 64; kk += 4) {
#pragma unroll
      for (int t = 0; t < 3; ++t) {
        v2f a;
        a.x = ap[(size_t)(ks + kk + kh) * 3 + t];
        a.y = ap[(size_t)(ks + kk + kh + 1) * 3 + t];
#pragma unroll
        for (int j = 0; j < 4; ++j) {
          const int nl = j * 16 + lm + t + 1;  // (n0 + j*16+lm + t-1) - (n0-2)
          v2f bf;
          bf.x = bs[nl * 68 + kk + kh];
          bf.y = bs[nl * 68 + kk + kh + 1];
          acc[j] = wmma4(a, bf, acc[j]);
        }
      }
    }
    __syncthreads();
  }
  float* out = xi + (size_t)b * DINNER * LSEQ;
#pragma unroll
  for (int j = 0; j < 4; ++j)
#pragma unroll
    for (int r = 0; r < 8; ++r) {
      const int m = m0 + r + half * 8;
      out[(size_t)m * LSEQ + n0 + j * 16 + lm] = silu_f(acc[j][r] + bc[m]);
    }
}

// ---------------------------------------------------------------------------
// K3: x_dbl[b,o,l] = sum_d W[o,d] * xi[b,d,l]   (M=160, one wave per block)
// grid (LSEQ/64, 160/16, B), block 32.
// ---------------------------------------------------------------------------
__global__ __launch_bounds__(32)
void k_xproj(const float* __restrict__ xi, const float* __restrict__ w,
             float* __restrict__ xdbl) {
  const int lane = threadIdx.x & 31;
  const int lm   = lane & 15;
  const int half = lane >> 4;
  const int kh   = half * 2;
  const int b  = blockIdx.z;
  const int n0 = blockIdx.x * 64;
  const int m0 = blockIdx.y * 16;

  const float* src  = xi + (size_t)b * DINNER * LSEQ;
  const float* arow = w + (size_t)(m0 + lm) * DINNER + kh;
  v8f acc[4] = {};
  for (int k0 = 0; k0 < DINNER; k0 += 4) {
    v2f a; a.x = arow[0]; a.y = arow[1]; arow += 4;
#pragma unroll
    for (int j = 0; j < 4; ++j) {
      v2f bf;
      bf.x = src[(size_t)(k0 + kh) * LSEQ + n0 + j * 16 + lm];
      bf.y = src[(size_t)(k0 + kh + 1) * LSEQ + n0 + j * 16 + lm];
      acc[j] = wmma4(a, bf, acc[j]);
    }
  }
  float* out = xdbl + (size_t)b * XDBLR * LSEQ;
#pragma unroll
  for (int j = 0; j < 4; ++j)
#pragma unroll
    for (int r = 0; r < 8; ++r)
      out[(size_t)(m0 + r + half * 8) * LSEQ + n0 + j * 16 + lm] = acc[j][r];
}

// ---------------------------------------------------------------------------
// K4: fused dt-projection + softplus + selective scan.
// Half-wave (16 lanes) per (b,d,dir) channel; lane = SSM state n.
// Width-16 shuffles broadcast delta/u; tree reduction sums h*C over states.
// grid (B*DINNER*2/16), block 256.
// ---------------------------------------------------------------------------
__global__ __launch_bounds__(256)
void k_scan(const float* __restrict__ xi, const float* __restrict__ xdbl,
            const float* __restrict__ dtw, const float* __restrict__ dtb,
            const float* __restrict__ alog, const float* __restrict__ dvec,
            float* __restrict__ y0, float* __restrict__ y1) {
  const int lm = threadIdx.x & 15;                 // state index n
  const int g  = blockIdx.x * 16 + (threadIdx.x >> 4);
  const int dir = g / (BATCH * DINNER);
  const int rem = g - dir * (BATCH * DINNER);
  const int b = rem / DINNER;
  const int d = rem - b * DINNER;

  const float An   = -__expf(alog[((size_t)dir * DINNER + d) * DSTATE + lm]);
  const float Dv   = dvec[dir * DINNER + d];
  const float dtbv = dtb[dir * DINNER + d];
  const float* wdt = dtw + ((size_t)dir * DINNER + d) * DTRANK;
  const int ud = dir ? (DINNER - 1 - d) : d;       // reference flips channels for dir 1
  const float* urow = xi + ((size_t)b * DINNER + ud) * LSEQ;
  const float* xdb  = xdbl + (size_t)b * XDBLR * LSEQ;
  const float* dts  = xdb + (size_t)(dir * DTRANK) * LSEQ;
  const float* Brow = xdb + (size_t)(2 * DTRANK + dir * DSTATE + lm) * LSEQ;
  const float* Crow = xdb + (size_t)(2 * DTRANK + 2 * DSTATE + dir * DSTATE + lm) * LSEQ;
  float* ydst = (dir ? y1 : y0) + ((size_t)b * DINNER + d) * LSEQ;

  float h = 0.0f;
  for (int l0 = 0; l0 < LSEQ; l0 += 16) {
    const int l = l0 + lm;
    float draw = 0.0f;
#pragma unroll 8
    for (int r = 0; r < DTRANK; ++r)
      draw = fmaf(wdt[r], dts[(size_t)r * LSEQ + l], draw);
    draw += dtbv;
    const float delta = (draw > 20.0f) ? draw : log1pf(__expf(draw));
    const float uv = urow[l];

    float Bn[16], Cn[16];
#pragma unroll
    for (int s = 0; s < 16; ++s) { Bn[s] = Brow[l0 + s]; Cn[s] = Crow[l0 + s]; }

    float ymine = 0.0f;
#pragma unroll
    for (int s = 0; s < 16; ++s) {
      const float dl = __shfl(delta, s, 16);
      const float us = __shfl(uv, s, 16);
      const float dA = __expf(dl * An);
      h = fmaf(dA, h, dl * Bn[s] * us);          // h_l = exp(dt*A)*h + dt*B*u
      float p = h * Cn[s];
      p += __shfl_xor(p, 1, 16);
      p += __shfl_xor(p, 2, 16);
      p += __shfl_xor(p, 4, 16);
      p += __shfl_xor(p, 8, 16);                 // sum over 16 states
      if (lm == s) ymine = fmaf(uv, Dv, p);      // + u*D
    }
    ydst[l] = ymine;
  }
}

// ---------------------------------------------------------------------------
// K5: yg[b,d,l] = (y0 + y1) * silu(z);  written over xz rows 0..1535 (reuse)
// ---------------------------------------------------------------------------
__global__ __launch_bounds__(256)
void k_gate(const float* __restrict__ y0, const float* __restrict__ y1,
            float* __restrict__ xz) {
  const int idx = blockIdx.x * blockDim.x + threadIdx.x;
  if (idx >= BATCH * DINNER * LSEQ) return;
  const int b  = idx / (DINNER * LSEQ);
  const int rl = idx - b * (DINNER * LSEQ);
  float* base = xz + (size_t)b * 2 * DINNER * LSEQ;
  const float z = base[(size_t)DINNER * LSEQ + rl];
  base[rl] = (y0[idx] + y1[idx]) * silu_f(z);
}

// ---------------------------------------------------------------------------
// K6: out[b,l,m] = sum_d yg[b,d,l] * Wo[m,d]   (store transposed tile).
// Activation tile staged [n][k] via transposing async b32 copies.
// grid (LSEQ/64, DMODEL/128, B), block 256.
// ---------------------------------------------------------------------------
__global__ __launch_bounds__(256)
void k_outproj(const float* __restrict__ yg, const float* __restrict__ w,
               float* __restrict__ out) {
  __shared__ float bs[64 * 68];
  const int tid  = threadIdx.x;
  const int lane = tid & 31;
  const int wave = tid >> 5;
  const int lm   = lane & 15;
  const int half = lane >> 4;
  const int kh   = half * 2;
  const int b  = blockIdx.z;
  const int n0 = blockIdx.x * 64;
  const int m0 = blockIdx.y * 128 + wave * 16;

  const float* src  = yg + (size_t)b * 2 * DINNER * LSEQ;  // gated y in xz rows 0..1535
  const float* arow = w + (size_t)(m0 + lm) * DINNER + kh;
  const int skl = tid >> 2;         // staged k row (0..63)
  const int snb = (tid & 3) * 16;   // 16-column n chunk

  v8f acc[4] = {};
  for (int ks = 0; ks < DINNER; ks += 64) {
    const float* grow = src + (size_t)(ks + skl) * LSEQ + n0 + snb;
#ifdef HAVE_ASYNC_LDS
#pragma unroll
    for (int i = 0; i < 16; ++i)
      async_cp4(grow + i, &bs[(snb + i) * 68 + skl]);   // transposing stage
    async_wait0();
#else
#pragma unroll
    for (int i = 0; i < 16; ++i) bs[(snb + i) * 68 + skl] = grow[i];
#endif
    __syncthreads();
#pragma unroll
    for (int kk = 0; kk < 64; kk += 4) {
      if ((kk & 31) == 0) __builtin_prefetch(arow + 64, 0, 3);
      v2f a; a.x = arow[0]; a.y = arow[1]; arow += 4;
      const float* bb = &bs[lm * 68 + kk + kh];
#pragma unroll
      for (int j = 0; j < 4; ++j) {
        v2f bf; bf.x = bb[j * 16 * 68]; bf.y = bb[j * 16 * 68 + 1];
        acc[j] = wmma4(a, bf, acc[j]);
      }
    }
    __syncthreads();
  }
  float* ob = out + (size_t)b * LSEQ * DMODEL;
#pragma unroll
  for (int j = 0; j < 4; ++j)
#pragma unroll
    for (int r = 0; r < 8; ++r)
      ob[(size_t)(n0 + j * 16 + lm) * DMODEL + (m0 + r + half * 8)] = acc[j][r];
}

// ---------------------------------------------------------------------------
// TDM demo: stage a 16x16 f32 tile of in_proj_w into LDS via the Tensor Data
// Mover, wait on TENSORcnt, cluster barrier (NOP outside a cluster), consume.
// ---------------------------------------------------------------------------
#if __has_builtin(__builtin_amdgcn_tensor_load_to_lds)
typedef __attribute__((ext_vector_type(4))) unsigned int u32x4;
typedef __attribute__((ext_vector_type(4))) int i32x4;
typedef __attribute__((ext_vector_type(8))) int i32x8;

__global__ __launch_bounds__(32)
void k_tdm_warm(const float* __restrict__ w, float* __restrict__ sink) {
  __shared__ float tile[16 * 16];
  const unsigned long long ga = (unsigned long long)(const void*)w;
  const unsigned int lds = (unsigned int)(size_t)&tile[0];

  u32x4 g0 = {};
  g0.x = 1u;                                           // count=1 (valid user D#)
  g0.y = lds;                                          // lds_addr
  g0.z = (unsigned int)(ga & 0xFFFFFFFFu);             // global_addr[31:0]
  g0.w = (unsigned int)((ga >> 32) & 0x01FFFFFFu) | (2u << 30);  // addr[56:32], type=2

  i32x8 g1 = {};
  g1[0] = (int)(2u << 16);                             // data_size = 4 bytes
  g1[1] = (int)((DMODEL & 0xFFFFu) << 16);             // tensor_dim0 lo16 @bits63:48
  g1[2] = (int)((DMODEL >> 16) | ((3072u & 0xFFFFu) << 16));  // dim0 hi / dim1 lo
  g1[3] = (int)((3072u >> 16) | (16u << 16));          // dim1 hi / tile_dim0=16
  g1[4] = 16;                                          // tile_dim1=16, tile_dim2=0
  g1[5] = DMODEL;                                      // tensor_dim0_stride lo32
  g1[6] = 0; g1[7] = 0;

  i32x4 gz = {0, 0, 0, 0};
#if __clang_major__ >= 23
  i32x8 gz8 = {};
  __builtin_amdgcn_tensor_load_to_lds(g0, g1, gz, gz, gz8, 0);
#else
  __builtin_amdgcn_tensor_load_to_lds(g0, g1, gz, gz, 0);
#endif
  __builtin_amdgcn_s_wait_tensorcnt(0);
  __builtin_amdgcn_s_cluster_barrier();                // NOP when not in a cluster
  __syncthreads();
  sink[threadIdx.x] = tile[threadIdx.x];               // consume so LDS stays live
}
#endif

// ---------------------------------------------------------------------------
extern "C" void kernel_launch(void* const* d_in, const int* in_sizes, int n_in,
                              void* d_out, int out_size, void* d_ws, size_t ws_size,
                              hipStream_t stream) {
  (void)in_sizes; (void)n_in; (void)out_size; (void)ws_size;
  const float* x       = (const float*)d_in[0];
  const float* w_in    = (const float*)d_in[1];
  const float* w_conv  = (const float*)d_in[2];
  const float* b_conv  = (const float*)d_in[3];
  const float* w_xproj = (const float*)d_in[4];
  const float* w_dt    = (const float*)d_in[5];
  const float* b_dt    = (const float*)d_in[6];
  const float* a_logs  = (const float*)d_in[7];
  const float* dvec    = (const float*)d_in[8];
  const float* w_out   = (const float*)d_in[9];
  float* out = (float*)d_out;

  // Workspace layout (floats): xz | xi | xdbl | y0 | y1   (~64 MB total)
  float* xz   = (float*)d_ws;                              // B * 3072 * L
  float* xi   = xz   + (size_t)BATCH * 2 * DINNER * LSEQ;  // B * 1536 * L
  float* xdbl = xi   + (size_t)BATCH * DINNER * LSEQ;      // B * 160 * L
  float* y0   = xdbl + (size_t)BATCH * XDBLR * LSEQ;       // B * 1536 * L
  float* y1   = y0   + (size_t)BATCH * DINNER * LSEQ;      // B * 1536 * L

#if __has_builtin(__builtin_amdgcn_tensor_load_to_lds)
  k_tdm_warm<<<1, 32, 0, stream>>>(w_in, y1);  // y1 fully overwritten by k_scan
#endif
  k_in_proj<<<dim3(LSEQ / 64, (2 * DINNER) / 128, BATCH), 256, 0, stream>>>(x, w_in, xz);
  k_conv<<<dim3(LSEQ / 64, DINNER / 128, BATCH), 256, 0, stream>>>(xz, w_conv, b_conv, xi);
  k_xproj<<<dim3(LSEQ / 64, XDBLR / 16, BATCH), 32, 0, stream>>>(xi, w_xproj, xdbl);
  k_scan<<<(BATCH * DINNER * 2) / 16, 256, 0, stream>>>(xi, xdbl, w_dt, b_dt, a_logs,
                                                        dvec, y0, y1);
  k_gate<<<(BATCH * DINNER * LSEQ) / 256, 256, 0, stream>>>(y0, y1, xz);
  k_outproj<<<dim3(LSEQ / 64, DMODEL / 128, BATCH), 256, 0, stream>>>(xz, w_out, out);
}